// FRAP_73031623901897
// MI455X (gfx1250) — compile-verified
//
#include <hip/hip_runtime.h>

// ---------------------------------------------------------------------------
// FRAP head, compile-time-specialized to the reference's "only state[-1]
// matters" structure. Kernel 1 (1 wave) computes q_common / q_last (WMMA f32
// Q-head). Kernel 2 broadcasts 1 MB of output at store bandwidth.
// ---------------------------------------------------------------------------

typedef __attribute__((ext_vector_type(2))) float v2f;
typedef __attribute__((ext_vector_type(8))) float v8f;

__device__ __forceinline__ float frelu(float x) { return fmaxf(x, 0.0f); }

__launch_bounds__(32, 1)
__global__ void frap_head_kernel(
    const float* __restrict__ state, int nb,
    const float* __restrict__ dv_w1, const float* __restrict__ dv_b1,
    const float* __restrict__ dv_w2, const float* __restrict__ dv_b2,
    const float* __restrict__ dp_w1, const float* __restrict__ dp_b1,
    const float* __restrict__ dp_w2, const float* __restrict__ dp_b2,
    const float* __restrict__ emb_w, const float* __restrict__ emb_b,
    const float* __restrict__ cp_w1, const float* __restrict__ cp_b1,
    const float* __restrict__ cp_w2, const float* __restrict__ cp_b2,
    const float* __restrict__ cm_w1, const float* __restrict__ cm_b1,
    const float* __restrict__ cm_w2, const float* __restrict__ cm_b2,
    const float* __restrict__ cm_w3, const float* __restrict__ cm_b3,
    const float* __restrict__ cc_w1, const float* __restrict__ cc_b1,
    const float* __restrict__ cc_w2, const float* __restrict__ cc_b2,
    const float* __restrict__ q_w1,  const float* __restrict__ q_b1,
    const float* __restrict__ q_w2,  const float* __restrict__ q_b2,
    const float* __restrict__ q_w3,  const float* __restrict__ q_b3,
    const float* __restrict__ maskp,
    float* __restrict__ qpair /* out: [0..7]=q_common, [8..15]=q_last */)
{
  __shared__ float ss[16];
  __shared__ float xcat[8][8];
  __shared__ float dmat[8][16];
  __shared__ float pdm[8][16];
  __shared__ float zbuf[1792];      // (7,8,32) scatter order == raw (32,7,8) view
  __shared__ float t1[20][56];
  __shared__ float t2[20][56];
  __shared__ float y1[4][56];
  __shared__ float y2[20][56];
  __shared__ float y3[20][56];
  __shared__ float u1[8][56];
  __shared__ float zc2s[20];
  __shared__ float rcl[16];         // [0..7]=r_common, [8..15]=r_last
  __shared__ float hbuf[16][16];    // WMMA inter-layer staging

  const int lane = threadIdx.x;     // exactly one wave32

  // --- s = state[-1] ---
  if (lane < 16) ss[lane] = state[(size_t)(nb - 1) * 16 + lane];
  __syncthreads();

  // --- 16 tiny per-phase MLPs (1->2->4), one per lane ---
  if (lane < 16) {
    const int m = lane >> 3;            // 0: dv on s[:8], 1: dp on s[8:16]
    const int p = lane & 7;
    const float* w1 = m ? dp_w1 : dv_w1;
    const float* b1 = m ? dp_b1 : dv_b1;
    const float* w2 = m ? dp_w2 : dv_w2;
    const float* b2 = m ? dp_b2 : dv_b2;
    const float x  = ss[m * 8 + p];
    const float h0 = frelu(w1[p * 2 + 0] * x + b1[p * 2 + 0]);
    const float h1 = frelu(w1[p * 2 + 1] * x + b1[p * 2 + 1]);
#pragma unroll
    for (int o = 0; o < 4; ++o)
      xcat[p][m * 4 + o] =
          frelu(w2[p * 8 + o * 2 + 0] * h0 + w2[p * 8 + o * 2 + 1] * h1 + b2[p * 4 + o]);
  }
  __syncthreads();

  // --- d = relu(xcat @ emb_w.T + emb_b)  (8,16) ---
  for (int idx = lane; idx < 128; idx += 32) {
    const int p = idx >> 4, o = idx & 15;
    float acc = emb_b[o];
#pragma unroll
    for (int k = 0; k < 8; ++k) acc += xcat[p][k] * emb_w[o * 8 + k];
    dmat[p][o] = frelu(acc);
  }
  __syncthreads();

  // --- phase-pair demand ---
  {
    const int pa[8] = {0, 0, 4, 1, 2, 2, 6, 3};
    const int pb[8] = {4, 1, 5, 5, 6, 3, 7, 7};
    for (int idx = lane; idx < 128; idx += 32) {
      const int p = idx >> 4, o = idx & 15;
      pdm[p][o] = dmat[pa[p]][o] + dmat[pb[p]][o];
    }
  }
  __syncthreads();

  // --- scatter pair vectors into (7,8,32); raw view -> channels-first (32,7,8)
  for (int idx = lane; idx < 1792; idx += 32) {
    const int e = idx >> 5, c = idx & 31;
    const int j = e / 7, t = e % 7;
    const int i = (t < j) ? t : t + 1;  // ROWS=t, COLS=j, II=i, JJ=j
    zbuf[(t * 8 + j) * 32 + c] = (c < 16) ? pdm[i][c] : pdm[j][c - 16];
  }
  __syncthreads();

  // --- cp convs on last-batch tile: zin[c][hw] = zbuf[c*56+hw] ---
  for (int idx = lane; idx < 1120; idx += 32) {
    const int o = idx / 56, hw = idx % 56;
    float acc = cp_b1[o];
    for (int c = 0; c < 32; ++c) acc += cp_w1[o * 32 + c] * zbuf[c * 56 + hw];
    t1[o][hw] = frelu(acc);
  }
  __syncthreads();
  for (int idx = lane; idx < 1120; idx += 32) {
    const int o = idx / 56, hw = idx % 56;
    float acc = cp_b2[o];
    for (int c = 0; c < 20; ++c) acc += cp_w2[o * 20 + c] * t1[c][hw];
    t2[o][hw] = frelu(acc);
  }
  // common path (all-zero input => spatially-uniform channel values)
  if (lane < 20) {
    float acc = cp_b2[lane];
    for (int c = 0; c < 20; ++c) acc += cp_w2[lane * 20 + c] * frelu(cp_b1[c]);
    zc2s[lane] = frelu(acc);
  }
  // --- mask path y ---
  for (int idx = lane; idx < 224; idx += 32) {
    const int o = idx / 56, hw = idx % 56;
    y1[o][hw] = frelu(cm_w1[o] * maskp[hw] + cm_b1[o]);
  }
  __syncthreads();
  for (int idx = lane; idx < 1120; idx += 32) {
    const int o = idx / 56, hw = idx % 56;
    float acc = cm_b2[o];
#pragma unroll
    for (int c = 0; c < 4; ++c) acc += cm_w2[o * 4 + c] * y1[c][hw];
    y2[o][hw] = frelu(acc);
  }
  __syncthreads();
  for (int idx = lane; idx < 1120; idx += 32) {
    const int o = idx / 56, hw = idx % 56;
    float acc = cm_b3[o];
    for (int c = 0; c < 20; ++c) acc += cm_w3[o * 20 + c] * y2[c][hw];
    y3[o][hw] = frelu(acc);
  }
  __syncthreads();

  // --- cc convs + h-sum, for common (v=0) and last (v=1) rows ---
  for (int v = 0; v < 2; ++v) {
    for (int idx = lane; idx < 448; idx += 32) {
      const int o = idx / 56, hw = idx % 56;
      float acc = cc_b1[o];
      for (int c = 0; c < 20; ++c) {
        const float rch = (v ? t2[c][hw] : zc2s[c]) * y3[c][hw];
        acc += cc_w1[o * 20 + c] * rch;
      }
      u1[o][hw] = frelu(acc);
    }
    __syncthreads();
    if (lane < 8) {
      float rsum = 0.0f;
      for (int h = 0; h < 7; ++h) {
        const int hw = h * 8 + lane;
        float acc = cc_b2[0];
#pragma unroll
        for (int c = 0; c < 8; ++c) acc += cc_w2[c] * u1[c][hw];
        rsum += frelu(acc);
      }
      rcl[v * 8 + lane] = rsum;
    }
    __syncthreads();
  }

  // ========================================================================
  // Q-head as a 16x16 fp32 tile via V_WMMA_F32_16X16X4_F32 (exact fp32).
  // Rows 0/1 of A are r_common / r_last; the rest are don't-care padding.
  // Layouts (ISA 7.12.2): A 16x4: lane holds M=lane&15, K = {0,1}|{2,3} per
  // lane half; B mirrored (N=lane&15); C/D: c[v] = C[v+8*(lane>>4)][lane&15].
  // ========================================================================
  const int mn    = lane & 15;
  const int khalf = (lane >> 4) * 2;

  // Layer 1: (16x8) @ W1^T(8x16), W1T[k][n] = q_w1[n*8+k]
  v8f acc1 = {0.f, 0.f, 0.f, 0.f, 0.f, 0.f, 0.f, 0.f};
#pragma unroll
  for (int k0 = 0; k0 < 8; k0 += 4) {
    const int ka = k0 + khalf;
    v2f a, b;
    a.x = (mn < 2) ? rcl[mn * 8 + ka] : 0.f;
    a.y = (mn < 2) ? rcl[mn * 8 + ka + 1] : 0.f;
    b.x = q_w1[mn * 8 + ka];
    b.y = q_w1[mn * 8 + ka + 1];
    acc1 = __builtin_amdgcn_wmma_f32_16x16x4_f32(false, a, false, b, (short)0,
                                                 acc1, false, false);
  }
  {
    const float bn = q_b1[mn];
#pragma unroll
    for (int v = 0; v < 8; ++v)
      hbuf[v + 8 * (lane >> 4)][mn] = frelu(acc1[v] + bn);
  }
  __syncthreads();

  // Layer 2: (16x16) @ W2^T(16x16), W2T[k][n] = q_w2[n*16+k]
  v8f acc2 = {0.f, 0.f, 0.f, 0.f, 0.f, 0.f, 0.f, 0.f};
#pragma unroll
  for (int k0 = 0; k0 < 16; k0 += 4) {
    const int ka = k0 + khalf;
    v2f a, b;
    a.x = hbuf[mn][ka];
    a.y = hbuf[mn][ka + 1];
    b.x = q_w2[mn * 16 + ka];
    b.y = q_w2[mn * 16 + ka + 1];
    acc2 = __builtin_amdgcn_wmma_f32_16x16x4_f32(false, a, false, b, (short)0,
                                                 acc2, false, false);
  }
  __syncthreads();  // hbuf reads done; safe to overwrite
  {
    const float bn = q_b2[mn];
#pragma unroll
    for (int v = 0; v < 8; ++v)
      hbuf[v + 8 * (lane >> 4)][mn] = frelu(acc2[v] + bn);
  }
  __syncthreads();

  // Layer 3: (16x16) @ W3^T(16x8 zero-padded to 16), W3T[k][n] = q_w3[n*16+k]
  v8f acc3 = {0.f, 0.f, 0.f, 0.f, 0.f, 0.f, 0.f, 0.f};
#pragma unroll
  for (int k0 = 0; k0 < 16; k0 += 4) {
    const int ka = k0 + khalf;
    v2f a, b;
    a.x = hbuf[mn][ka];
    a.y = hbuf[mn][ka + 1];
    b.x = (mn < 8) ? q_w3[mn * 16 + ka] : 0.f;
    b.y = (mn < 8) ? q_w3[mn * 16 + ka + 1] : 0.f;
    acc3 = __builtin_amdgcn_wmma_f32_16x16x4_f32(false, a, false, b, (short)0,
                                                 acc3, false, false);
  }

  // Extract rows 0 (common) and 1 (last): lanes 0..7 hold them in c[0]/c[1].
  if (lane < 8) {
    const float bn = q_b3[lane];
    qpair[lane]     = frelu(acc3[0] + bn);
    qpair[lane + 8] = frelu(acc3[1] + bn);
  }
}

// ---------------------------------------------------------------------------
// Broadcast: rows 0..nb-2 <- q_common, row nb-1 <- q_last. 2 float4 per row.
// Pure store-bandwidth kernel (1 MB).
// ---------------------------------------------------------------------------
__global__ void frap_bcast_kernel(const float4* __restrict__ qpair4,
                                  float4* __restrict__ out, int nrow) {
  const int f = blockIdx.x * blockDim.x + threadIdx.x;  // float4 index
  if (f >= nrow * 2) return;
  const int row = f >> 1, half = f & 1;
  out[f] = qpair4[((row == nrow - 1) ? 2 : 0) + half];
}

extern "C" void kernel_launch(void* const* d_in, const int* in_sizes, int n_in,
                              void* d_out, int out_size, void* d_ws, size_t ws_size,
                              hipStream_t stream) {
  const float* state = (const float*)d_in[0];
  const float* dv_w1 = (const float*)d_in[1];
  const float* dv_b1 = (const float*)d_in[2];
  const float* dv_w2 = (const float*)d_in[3];
  const float* dv_b2 = (const float*)d_in[4];
  const float* dp_w1 = (const float*)d_in[5];
  const float* dp_b1 = (const float*)d_in[6];
  const float* dp_w2 = (const float*)d_in[7];
  const float* dp_b2 = (const float*)d_in[8];
  const float* emb_w = (const float*)d_in[9];
  const float* emb_b = (const float*)d_in[10];
  const float* cp_w1 = (const float*)d_in[11];
  const float* cp_b1 = (const float*)d_in[12];
  const float* cp_w2 = (const float*)d_in[13];
  const float* cp_b2 = (const float*)d_in[14];
  const float* cm_w1 = (const float*)d_in[15];
  const float* cm_b1 = (const float*)d_in[16];
  const float* cm_w2 = (const float*)d_in[17];
  const float* cm_b2 = (const float*)d_in[18];
  const float* cm_w3 = (const float*)d_in[19];
  const float* cm_b3 = (const float*)d_in[20];
  const float* cc_w1 = (const float*)d_in[21];
  const float* cc_b1 = (const float*)d_in[22];
  const float* cc_w2 = (const float*)d_in[23];
  const float* cc_b2 = (const float*)d_in[24];
  const float* q_w1  = (const float*)d_in[25];
  const float* q_b1  = (const float*)d_in[26];
  const float* q_w2  = (const float*)d_in[27];
  const float* q_b2  = (const float*)d_in[28];
  const float* q_w3  = (const float*)d_in[29];
  const float* q_b3  = (const float*)d_in[30];
  const float* maskp = (const float*)d_in[31];

  const int nb = in_sizes[0] / 16;       // batch rows
  float* qpair = (float*)d_ws;           // 16 floats of scratch

  frap_head_kernel<<<1, 32, 0, stream>>>(
      state, nb, dv_w1, dv_b1, dv_w2, dv_b2, dp_w1, dp_b1, dp_w2, dp_b2,
      emb_w, emb_b, cp_w1, cp_b1, cp_w2, cp_b2, cm_w1, cm_b1, cm_w2, cm_b2,
      cm_w3, cm_b3, cc_w1, cc_b1, cc_w2, cc_b2, q_w1, q_b1, q_w2, q_b2,
      q_w3, q_b3, maskp, qpair);

  const int total4 = nb * 2;             // float4 elements of output
  frap_bcast_kernel<<<(total4 + 255) / 256, 256, 0, stream>>>(
      (const float4*)qpair, (float4*)d_out, nb);
}